// RiemannianAutoencoder_57604101374758
// MI455X (gfx1250) — compile-verified
//
#include <hip/hip_runtime.h>
#include <math.h>

// CDNA5 / gfx1250: wave32, WMMA f32 16x16x4
typedef __attribute__((ext_vector_type(2))) float v2f;
typedef __attribute__((ext_vector_type(8))) float v8f;

#define CBOX   4.0f
#define TSTEPS 50
#define BPART  8192
#define DTW    (1.0f/49.0f)
#define PI_F   3.14159265358979323846f
#define W0F    (PI_F/(2.0f*CBOX))   // pi/(2L), L == C == 4

// Geodesic RHS acceleration for one particle (duplicated in lanes l and l+16).
// af[d][kc][vg] : A-matrix fragments of Bsc_d = sqrt(spd_d)*beta_d / L, laid out
// for V_WMMA_F32_16X16X4_F32 (M=j0=lane%16, K=j1 chunked by 4).
__device__ __forceinline__ void rhs_acc(
    float x0, float x1, float v0, float v1,
    const float af[3][4][2], int hi,
    float& a0, float& a1)
{
    const float alpha0 = W0F * (x0 + CBOX);
    const float alpha1 = W0F * (x1 + CBOX);

    // B fragments: per-particle sine vectors along j1 (this lane's K subset)
    float bs1[8], bt1[8];
#pragma unroll
    for (int kc = 0; kc < 4; ++kc) {
#pragma unroll
        for (int vg = 0; vg < 2; ++vg) {
            const int j1 = 4 * kc + 2 * hi + vg + 1;   // 1-based frequency
            float s, c;
            __sincosf((float)j1 * alpha1, &s, &c);
            bs1[2 * kc + vg] = s;                       // for phi and dphi/dx0
            bt1[2 * kc + vg] = (W0F * (float)j1) * c;   // for dphi/dx1
        }
    }

    // s0 / w0*c0 along j0 for this lane's half of the D rows
    float s0v[8], g0v[8];
#pragma unroll
    for (int r = 0; r < 8; ++r) {
        const int j0 = r + 8 * hi + 1;
        float s, c;
        __sincosf((float)j0 * alpha0, &s, &c);
        s0v[r] = s;
        g0v[r] = (W0F * (float)j0) * c;
    }

    float f[3], df0[3], df1[3];
#pragma unroll
    for (int d = 0; d < 3; ++d) {
        v8f U = {0.f, 0.f, 0.f, 0.f, 0.f, 0.f, 0.f, 0.f}; // Bsc_d @ s1
        v8f V = {0.f, 0.f, 0.f, 0.f, 0.f, 0.f, 0.f, 0.f}; // Bsc_d @ (w1*c1)
#pragma unroll
        for (int kc = 0; kc < 4; ++kc) {
            v2f a;  a.x  = af[d][kc][0]; a.y  = af[d][kc][1];
            v2f bs; bs.x = bs1[2 * kc];  bs.y = bs1[2 * kc + 1];
            v2f bt; bt.x = bt1[2 * kc];  bt.y = bt1[2 * kc + 1];
            U = __builtin_amdgcn_wmma_f32_16x16x4_f32(false, a, false, bs,
                                                      (short)0, U, false, false);
            V = __builtin_amdgcn_wmma_f32_16x16x4_f32(false, a, false, bt,
                                                      (short)0, V, false, false);
        }
        // D layout: VGPR r holds row M = r + 8*hi, column N = lane%16 (our particle)
        float fp = 0.f, d0p = 0.f, d1p = 0.f;
#pragma unroll
        for (int r = 0; r < 8; ++r) {
            fp  += s0v[r] * U[r];   // f_d      = s0 . U_d
            d0p += g0v[r] * U[r];   // df_d/dx0 = (w0*c0) . U_d
            d1p += s0v[r] * V[r];   // df_d/dx1 = s0 . V_d
        }
        f[d]   = fp  + __shfl_xor(fp,  16, 32);
        df0[d] = d0p + __shfl_xor(d0p, 16, 32);
        df1[d] = d1p + __shfl_xor(d1p, 16, 32);
    }

    // metric g = I + [[f0, f1],[f1, f2]];  dg_ij/dx_n from df
    const float g00 = 1.f + f[0], g01 = f[1], g11 = 1.f + f[2];
    const float det = g00 * g11 - g01 * g01;
    const float inv = 1.f / det;
    const float i00 = g11 * inv, i01 = -g01 * inv, i11 = g00 * inv;

    // Dg_lj = v^i d_i g_lj
    const float Dg00 = v0 * df0[0] + v1 * df1[0];
    const float Dg01 = v0 * df0[1] + v1 * df1[1];
    const float Dg11 = v0 * df0[2] + v1 * df1[2];
    const float h0 = Dg00 * v0 + Dg01 * v1;
    const float h1 = Dg01 * v0 + Dg11 * v1;
    // w_n = d_n (v^T G v)
    const float w0 = v0 * v0 * df0[0] + 2.f * v0 * v1 * df0[1] + v1 * v1 * df0[2];
    const float w1 = v0 * v0 * df1[0] + 2.f * v0 * v1 * df1[1] + v1 * v1 * df1[2];
    const float Q0 = 2.f * h0 - w0;
    const float Q1 = 2.f * h1 - w1;
    a0 = -0.5f * (i00 * Q0 + i01 * Q1);
    a1 = -0.5f * (i01 * Q0 + i11 * Q1);
}

__global__ void __launch_bounds__(32)
geodesic_rk4_kernel(const float* __restrict__ ic,    // [8192, 4]
                    const float* __restrict__ beta,  // [3, 256]
                    const float* __restrict__ ls,    // [3, 2]
                    float* __restrict__ out)         // [50, 8192, 2]
{
    // One wave per workgroup: 512 independent waves spread across WGPs.
    const int lane = threadIdx.x & 31;
    const int hi   = lane >> 4;
    const int plo  = lane & 15;
    const int p    = blockIdx.x * 16 + plo;           // particle id (dup in both halves)

    // --- fold beta, ls into WMMA A-fragments of Bsc_d (constant for whole run) ---
    float af[3][4][2];
    const int   j0  = plo + 1;
    const float om0 = W0F * (float)j0;
#pragma unroll
    for (int d = 0; d < 3; ++d) {
        const float ls0 = ls[d * 2 + 0];
        const float ls1 = ls[d * 2 + 1];
        const float pre = 2.0f * PI_F * ls0 * ls1;    // (2*pi)^(N/2) * prod(ls), N=2
        const float t0  = ls0 * om0;
#pragma unroll
        for (int kc = 0; kc < 4; ++kc) {
#pragma unroll
            for (int vg = 0; vg < 2; ++vg) {
                const int   j1  = 4 * kc + 2 * hi + vg + 1;
                const float om1 = W0F * (float)j1;
                const float t1  = ls1 * om1;
                const float spd = pre * __expf(-0.5f * (t0 * t0 + t1 * t1));
                const int   m   = (j0 - 1) * 16 + (j1 - 1);
                af[d][kc][vg] = sqrtf(spd) * beta[d * 256 + m] * (1.0f / CBOX);
            }
        }
    }

    // --- load state (both halves load the same particle; coalesced float4) ---
    const float4 s0 = reinterpret_cast<const float4*>(ic)[p];
    float x0 = s0.x, x1 = s0.y, v0 = s0.z, v1 = s0.w;

    float2* outp = reinterpret_cast<float2*>(out);
    if (!hi) outp[p] = make_float2(x0, x1);           // t = 0 row

    auto stage = [&](float cx0, float cx1, float cv0, float cv1,
                     float& kx0, float& kx1, float& kv0, float& kv1) {
        kx0 = cv0; kx1 = cv1;
        rhs_acc(cx0, cx1, cv0, cv1, af, hi, kv0, kv1);
    };

    for (int t = 1; t < TSTEPS; ++t) {
        float k1x0, k1x1, k1v0, k1v1;
        float k2x0, k2x1, k2v0, k2v1;
        float k3x0, k3x1, k3v0, k3v1;
        float k4x0, k4x1, k4v0, k4v1;

        stage(x0, x1, v0, v1, k1x0, k1x1, k1v0, k1v1);
        stage(x0 + 0.5f * DTW * k1x0, x1 + 0.5f * DTW * k1x1,
              v0 + 0.5f * DTW * k1v0, v1 + 0.5f * DTW * k1v1,
              k2x0, k2x1, k2v0, k2v1);
        stage(x0 + 0.5f * DTW * k2x0, x1 + 0.5f * DTW * k2x1,
              v0 + 0.5f * DTW * k2v0, v1 + 0.5f * DTW * k2v1,
              k3x0, k3x1, k3v0, k3v1);
        stage(x0 + DTW * k3x0, x1 + DTW * k3x1,
              v0 + DTW * k3v0, v1 + DTW * k3v1,
              k4x0, k4x1, k4v0, k4v1);

        const float c6 = DTW / 6.0f;
        x0 += c6 * (k1x0 + 2.f * k2x0 + 2.f * k3x0 + k4x0);
        x1 += c6 * (k1x1 + 2.f * k2x1 + 2.f * k3x1 + k4x1);
        v0 += c6 * (k1v0 + 2.f * k2v0 + 2.f * k3v0 + k4v0);
        v1 += c6 * (k1v1 + 2.f * k2v1 + 2.f * k3v1 + k4v1);

        if (!hi) outp[(size_t)t * BPART + p] = make_float2(x0, x1);
    }
}

extern "C" void kernel_launch(void* const* d_in, const int* in_sizes, int n_in,
                              void* d_out, int out_size, void* d_ws, size_t ws_size,
                              hipStream_t stream) {
    (void)in_sizes; (void)n_in; (void)out_size; (void)d_ws; (void)ws_size;
    const float* ic   = (const float*)d_in[0];  // [8192, 4]
    const float* beta = (const float*)d_in[1];  // [3, 256]
    const float* ls   = (const float*)d_in[2];  // [3, 2]
    float* out        = (float*)d_out;          // [50, 8192, 2]

    // 8192 particles, 16 per wave32 -> 512 single-wave workgroups,
    // maximizing WGP spread for this VALU/matrix-bound integrator.
    geodesic_rk4_kernel<<<dim3(512), dim3(32), 0, stream>>>(ic, beta, ls, out);
}